// ProbabilisticSurfaceDistanceLoss_39608188404355
// MI455X (gfx1250) — compile-verified
//
#include <hip/hip_runtime.h>

// ---------------------------------------------------------------------------
// Probabilistic surface distance loss on gfx1250 (MI455X).
// d(i,j) = |s_i|^2 + |o_j|^2 - 2 s_i.o_j  folded into one f32 WMMA per tile:
//   A row  = (-2sx, -2sy, -2sz, 1)            (16x4, K=4)
//   B col  = ( ox,   oy,   oz,  |o|^2 )       (4x16)
//   A*B    = -2 s.o + |o|^2 ; |s|^2 added after the min (row-constant).
// Each wave carries 5 simp tiles (A fragments) and double-buffers B tiles,
// folding two B tiles per v_min3_num_f32 into the running minima.
// ---------------------------------------------------------------------------

typedef __attribute__((ext_vector_type(2))) float v2f;
typedef __attribute__((ext_vector_type(8))) float v8f;

#define F_ORIG   50000
#define F_SIMP   10000
#define ST_TILES 625      // 10000 / 16
#define OT_TILES 3125     // 50000 / 16
#define NSPLIT   25       // orig-dimension chunks
#define TPC      125      // tiles per chunk = 3125 / 25
#define SPW      5        // simp tiles per wave (625 = 5 * 125)
#define SG       125      // simp groups = ST_TILES / SPW

// ---- barycenter + extension precompute -----------------------------------

__global__ void bary_orig_kernel(const float* __restrict__ verts,
                                 const int* __restrict__ faces,
                                 float4* __restrict__ origExt, int nf) {
    int i = blockIdx.x * blockDim.x + threadIdx.x;
    if (i >= nf) return;
    int i0 = faces[3 * i + 0], i1 = faces[3 * i + 1], i2 = faces[3 * i + 2];
    const float inv3 = 1.0f / 3.0f;
    float bx = (verts[3 * i0 + 0] + verts[3 * i1 + 0] + verts[3 * i2 + 0]) * inv3;
    float by = (verts[3 * i0 + 1] + verts[3 * i1 + 1] + verts[3 * i2 + 1]) * inv3;
    float bz = (verts[3 * i0 + 2] + verts[3 * i1 + 2] + verts[3 * i2 + 2]) * inv3;
    origExt[i] = make_float4(bx, by, bz, bx * bx + by * by + bz * bz);
}

__global__ void bary_simp_kernel(const float* __restrict__ verts,
                                 const int* __restrict__ faces,
                                 float4* __restrict__ simpA,
                                 float* __restrict__ s2, int nf) {
    int i = blockIdx.x * blockDim.x + threadIdx.x;
    if (i >= nf) return;
    int i0 = faces[3 * i + 0], i1 = faces[3 * i + 1], i2 = faces[3 * i + 2];
    const float inv3 = 1.0f / 3.0f;
    float bx = (verts[3 * i0 + 0] + verts[3 * i1 + 0] + verts[3 * i2 + 0]) * inv3;
    float by = (verts[3 * i0 + 1] + verts[3 * i1 + 1] + verts[3 * i2 + 1]) * inv3;
    float bz = (verts[3 * i0 + 2] + verts[3 * i1 + 2] + verts[3 * i2 + 2]) * inv3;
    simpA[i] = make_float4(-2.0f * bx, -2.0f * by, -2.0f * bz, 1.0f);
    s2[i]    = bx * bx + by * by + bz * bz;
}

// ---- main: one wave per (5 simp tiles, orig chunk) ------------------------

__global__ __launch_bounds__(32) void
tile_min_kernel(const float2* __restrict__ simpA2,   // simpA viewed as float2[2] per row
                const float2* __restrict__ origE2,   // origExt viewed as float2[2] per col
                float* __restrict__ partial) {
    const int lane = threadIdx.x;          // wave32
    const int half = lane >> 4;            // K-slice: 0 -> K=0,1 ; 1 -> K=2,3
    const int idx  = lane & 15;            // M (A) / N (B,C) index
    const int sg   = blockIdx.x;           // simp group (5 consecutive tiles)
    const int oc   = blockIdx.y;           // orig chunk

    // A fragments: lane holds A[idx][2*half + {0,1}] for each of 5 simp tiles
    v2f a[SPW];
#pragma unroll
    for (int s = 0; s < SPW; ++s) {
        float2 af = simpA2[((sg * SPW + s) * 16 + idx) * 2 + half];
        a[s][0] = af.x; a[s][1] = af.y;
    }

    const float INF = __builtin_inff();
    float rm[SPW][8];
#pragma unroll
    for (int s = 0; s < SPW; ++s)
#pragma unroll
        for (int v = 0; v < 8; ++v) rm[s][v] = INF;

    // B stream for this chunk; one tile = 16 float4 = 32 float2 per lane-view.
    const float2* bp = origE2 + ((size_t)(oc * TPC) * 16 + idx) * 2 + half;

    // Double-buffered prologue: tiles 0 and 1.
    float2 b0f = bp[0];
    float2 b1f = bp[32];
    bp += 64;

    // 62 rounds of 2 tiles (0..123), epilogue handles tile 124.
    for (int i = 0; i < (TPC - 1) / 2; ++i) {
        // Prefetch next pair. Final round touches one tile past the chunk
        // (or, for the last chunk, 256B past origExt but still inside d_ws);
        // that value is never consumed.
        float2 n0 = bp[0];
        float2 n1 = bp[32];
        bp += 64;

        v2f b0; b0[0] = b0f.x; b0[1] = b0f.y;
        v2f b1; b1[0] = b1f.x; b1[1] = b1f.y;
#pragma unroll
        for (int s = 0; s < SPW; ++s) {
            v8f z0 = {0.f, 0.f, 0.f, 0.f, 0.f, 0.f, 0.f, 0.f};
            v8f z1 = {0.f, 0.f, 0.f, 0.f, 0.f, 0.f, 0.f, 0.f};
            v8f c0 = __builtin_amdgcn_wmma_f32_16x16x4_f32(
                         false, a[s], false, b0, (short)0, z0, false, false);
            v8f c1 = __builtin_amdgcn_wmma_f32_16x16x4_f32(
                         false, a[s], false, b1, (short)0, z1, false, false);
#pragma unroll
            for (int v = 0; v < 8; ++v) {
                // rm = min(rm, c0, c1) in one VALU op, no canonicalize junk.
                asm("v_min3_num_f32 %0, %0, %1, %2"
                    : "+v"(rm[s][v]) : "v"(c0[v]), "v"(c1[v]));
            }
        }
        b0f = n0; b1f = n1;
    }

    // Epilogue: tile 124.
    {
        v2f b0; b0[0] = b0f.x; b0[1] = b0f.y;
#pragma unroll
        for (int s = 0; s < SPW; ++s) {
            v8f z0 = {0.f, 0.f, 0.f, 0.f, 0.f, 0.f, 0.f, 0.f};
            v8f c0 = __builtin_amdgcn_wmma_f32_16x16x4_f32(
                         false, a[s], false, b0, (short)0, z0, false, false);
#pragma unroll
            for (int v = 0; v < 8; ++v)
                asm("v_min_num_f32 %0, %0, %1" : "+v"(rm[s][v]) : "v"(c0[v]));
        }
    }

    // Min across the 16 columns of each lane-half (xor masks stay in-half).
#pragma unroll
    for (int m = 8; m >= 1; m >>= 1) {
#pragma unroll
        for (int s = 0; s < SPW; ++s)
#pragma unroll
            for (int v = 0; v < 8; ++v) {
                float o = __shfl_xor(rm[s][v], m, 32);
                asm("v_min_num_f32 %0, %0, %1" : "+v"(rm[s][v]) : "v"(o));
            }
    }

    // C layout: VGPR v, lane-half h -> row M = v + 8*h. Lanes 0 and 16 write.
    if (idx == 0) {
#pragma unroll
        for (int s = 0; s < SPW; ++s)
#pragma unroll
            for (int v = 0; v < 8; ++v)
                partial[((sg * SPW + s) * 16 + v + 8 * half) * NSPLIT + oc] = rm[s][v];
    }
}

// ---- deterministic finalize: min over chunks, weighted fixed-tree sum -----

__global__ void finalize_kernel(const float* __restrict__ partial,
                                const float* __restrict__ s2,
                                const float* __restrict__ prob,
                                float* __restrict__ out) {
    __shared__ float red[256];
    int t = threadIdx.x;
    float acc = 0.0f;
    for (int row = t; row < F_SIMP; row += 256) {
        float m = partial[row * NSPLIT];
        for (int c = 1; c < NSPLIT; ++c)
            m = fminf(m, partial[row * NSPLIT + c]);
        acc += prob[row] * (s2[row] + m);   // add the row-constant |s|^2 here
    }
    red[t] = acc;
    __syncthreads();
    for (int s = 128; s > 0; s >>= 1) {
        if (t < s) red[t] += red[t + s];
        __syncthreads();
    }
    if (t == 0) out[0] = red[0];
}

// ---------------------------------------------------------------------------

extern "C" void kernel_launch(void* const* d_in, const int* in_sizes, int n_in,
                              void* d_out, int out_size, void* d_ws, size_t ws_size,
                              hipStream_t stream) {
    (void)in_sizes; (void)n_in; (void)out_size; (void)ws_size;

    const float* ov = (const float*)d_in[0];   // original_vertices   [30000,3]
    const int*   of = (const int*)  d_in[1];   // original_faces      [50000,3]
    const float* sv = (const float*)d_in[2];   // simplified_vertices [6000,3]
    const int*   sf = (const int*)  d_in[3];   // simplified_faces    [10000,3]
    const float* fp = (const float*)d_in[4];   // face_probabilities  [10000]

    char* ws = (char*)d_ws;
    float4* origExt = (float4*)(ws + 0);        // 50000 * 16 B = 800000
    float4* simpA   = (float4*)(ws + 800000);   // 10000 * 16 B = 160000
    float*  s2      = (float*) (ws + 960000);   // 10000 *  4 B =  40000
    float*  partial = (float*) (ws + 1000000);  // 10000 * 25 * 4 B = 1000000

    bary_orig_kernel<<<(F_ORIG + 255) / 256, 256, 0, stream>>>(ov, of, origExt, F_ORIG);
    bary_simp_kernel<<<(F_SIMP + 255) / 256, 256, 0, stream>>>(sv, sf, simpA, s2, F_SIMP);

    tile_min_kernel<<<dim3(SG, NSPLIT), 32, 0, stream>>>(
        (const float2*)simpA, (const float2*)origExt, partial);

    finalize_kernel<<<1, 256, 0, stream>>>(partial, s2, fp, (float*)d_out);
}